// Aggregator_20710332301461
// MI455X (gfx1250) — compile-verified
//
#include <hip/hip_runtime.h>

// out[i, :] = (1/32) * sum_j features[neighbor_idx[i, j], :]
// N_TARGETS = 100000, NUM_SAMPLES = 32, D = 128, features fp32.
//
// One wave32 per target. Reduction runs on the CDNA5 matrix pipe with
// V_WMMA_F32_16X16X4_F32 (exact fp32): A = uniform 1/32 matrix, B holds 4
// samples (K) x 16 channels (N), 8 chained WMMAs reduce all 32 samples.
// Since A is uniform, the sample->K-slot permutation inside B is irrelevant;
// only the N = lane&15 striping matters (documented B/C/D layout).
//
// Loop order: samples OUTER, channels INNER -> each random 512B feature row
// is fetched in one 16-load clause (full line + DRAM-row locality, single
// visit per row), with 8 independent WMMA accumulator chains absorbing it.

typedef __attribute__((ext_vector_type(2))) float v2f;
typedef __attribute__((ext_vector_type(8))) float v8f;

#define CH 128   // channels (D)
#define NS 32    // samples per target
#define WPB 8    // waves (targets) per 256-thread block

__global__ __launch_bounds__(256) void Aggregator_gather_mean_wmma(
    const float* __restrict__ features,   // [n_nodes, 128] fp32
    const int*   __restrict__ nidx,       // [n_targets, 32] int32
    float*       __restrict__ out,        // [n_targets, 128] fp32
    int n_targets)
{
    const int lane   = threadIdx.x & 31;
    const int wave   = threadIdx.x >> 5;
    const int target = blockIdx.x * WPB + wave;
    if (target >= n_targets) return;   // wave-uniform exit: EXEC stays all-1s

    // Lane l holds neighbor l's row index (coalesced 128B load per wave).
    const int my_idx = nidx[(size_t)target * NS + lane];

    const int half = (lane >> 4) & 1;  // 0: lanes 0-15, 1: lanes 16-31
    const int col  = lane & 15;        // N coordinate within a 16-ch block

    const v2f a = { 0.03125f, 0.03125f };   // A matrix: all entries 1/32 (exact pow2)

    // Accumulators for all 4 channel-pair blocks stay live across the sample
    // loop: accA[n] covers channels [32n, 32n+16), accB[n] covers [32n+16, 32n+32).
    v8f accA[4], accB[4];
    #pragma unroll
    for (int n = 0; n < 4; ++n) {
        accA[n] = (v8f){0.f,0.f,0.f,0.f,0.f,0.f,0.f,0.f};
        accB[n] = (v8f){0.f,0.f,0.f,0.f,0.f,0.f,0.f,0.f};
    }

    #pragma unroll
    for (int q = 0; q < 8; ++q) {
        // Broadcast this chunk's 4 sample row indices to SGPRs.
        const int r0 = __builtin_amdgcn_readlane(my_idx, 4*q + 0);
        const int r1 = __builtin_amdgcn_readlane(my_idx, 4*q + 1);
        const int r2 = __builtin_amdgcn_readlane(my_idx, 4*q + 2);
        const int r3 = __builtin_amdgcn_readlane(my_idx, 4*q + 3);
        // Lanes 0-15 carry samples {r0, r2}; lanes 16-31 carry {r1, r3}.
        // (Any distribution of 4 samples over B's 4 slots per column is valid
        //  because A is uniform.)
        const int rowx = half ? r1 : r0;
        const int rowy = half ? r3 : r2;
        const int ox = rowx * CH + col;   // <= 128M, fits int
        const int oy = rowy * CH + col;

        // One 16-load clause fetches the ENTIRE 512B of both rows:
        // 16 lanes x 4B x 8 offsets per row -> every byte once, contiguous.
        v2f bA[4], bB[4];
        #pragma unroll
        for (int n = 0; n < 4; ++n) {
            bA[n] = (v2f){ features[ox + 32*n],      features[oy + 32*n]      };
            bB[n] = (v2f){ features[ox + 32*n + 16], features[oy + 32*n + 16] };
        }

        #pragma unroll
        for (int n = 0; n < 4; ++n) {
            accA[n] = __builtin_amdgcn_wmma_f32_16x16x4_f32(
                          false, a, false, bA[n], (short)0, accA[n], false, false);
            accB[n] = __builtin_amdgcn_wmma_f32_16x16x4_f32(
                          false, a, false, bB[n], (short)0, accB[n], false, false);
        }
    }

    // C/D layout: lane L holds N = L&15 (replicated over M) in acc[0].
    // Lanes 0-15 emit block A, lanes 16-31 emit block B -> one coalesced
    // 128B store per channel-pair block.
    float* __restrict__ outp = out + (size_t)target * CH;
    #pragma unroll
    for (int n = 0; n < 4; ++n) {
        outp[32*n + lane] = half ? accB[n][0] : accA[n][0];
    }
}

extern "C" void kernel_launch(void* const* d_in, const int* in_sizes, int n_in,
                              void* d_out, int out_size, void* d_ws, size_t ws_size,
                              hipStream_t stream) {
    const float* features = (const float*)d_in[0];   // [n_nodes, 128] fp32
    const int*   nidx     = (const int*)d_in[1];     // [n_targets, 32] int32
    float*       out      = (float*)d_out;           // [n_targets, 128] fp32

    const int n_targets = in_sizes[1] / NS;
    const int blocks    = (n_targets + WPB - 1) / WPB;

    Aggregator_gather_mean_wmma<<<blocks, 256, 0, stream>>>(
        features, nidx, out, n_targets);
}